// ContextualViewModel_48833778155979
// MI455X (gfx1250) — compile-verified
//
#include <hip/hip_runtime.h>

typedef float v2f __attribute__((ext_vector_type(2)));
typedef float v8f __attribute__((ext_vector_type(8)));

#define H_  384
#define WG_ 384
#define F_  256
#define K_  256

#define LDS_STRIDE 260          // 256 + 4-float pad -> conflict-free ds_load_b64 of A fragments
#define MB_PER_BLOCK 16         // M 16-row blocks processed per workgroup
#define MB_TOTAL ((H_ * WG_) / 16)              // 9216
#define GRID_Y (MB_TOTAL / MB_PER_BLOCK)        // 576

// ---------------------------------------------------------------------------
// Kernel 1: yT[f][k] = sum_j x[sx[k], sy[k], j] * W[j][f]   (tiny, 33 MFLOP)
// Stored transposed so the main kernel's WMMA B fragments are contiguous b64.
// ---------------------------------------------------------------------------
__global__ __launch_bounds__(256) void station_gemm_kernel(
    const float* __restrict__ x, const float* __restrict__ W,
    const int* __restrict__ sx, const int* __restrict__ sy,
    float* __restrict__ yT)
{
    __shared__ float s[F_];
    const int k = blockIdx.x;    // station index
    const int f = threadIdx.x;   // output feature
    const float* row = x + ((size_t)sx[k] * WG_ + sy[k]) * F_;
    s[f] = row[f];
    __syncthreads();
    float acc = 0.f;
    #pragma unroll 8
    for (int j = 0; j < F_; ++j)
        acc = fmaf(s[j], W[j * F_ + f], acc);
    yT[f * K_ + k] = acc;        // transposed store (N-major)
}

// ---------------------------------------------------------------------------
// Async memory->LDS staging of one 16 x 256 fp32 d-tile (padded rows).
// Uses CDNA5 GLOBAL_LOAD_ASYNC_TO_LDS_B128 (ASYNCcnt-tracked, no VGPR bounce).
// ---------------------------------------------------------------------------
__device__ __forceinline__ void async_tile_load(const float* gbase, float* sbase, int tid)
{
    #pragma unroll
    for (int c = 0; c < 8; ++c) {
        const int idx = c * 128 + tid;          // float4 index within tile
        const int row = idx >> 6;               // 64 float4 per 256-float row
        const int col = (idx & 63) * 4;
        unsigned lds = (unsigned)(uintptr_t)(sbase + row * LDS_STRIDE + col);
        const float* g = gbase + (size_t)idx * 4;
        asm volatile("global_load_async_to_lds_b128 %0, %1, off"
                     :: "v"(lds), "v"(g) : "memory");
    }
}

__device__ __forceinline__ void wait_async_all()
{
    asm volatile("s_wait_asynccnt 0x0" ::: "memory");
}

// ---------------------------------------------------------------------------
// Kernel 2: out[m][n] = sum_k d[m][k] * y[k][n]   (M=147456, N=256, K=256)
// fp32 WMMA 16x16x4. B panel (16 N-cols x 256 K) register-resident per wave.
// Double-buffered async LDS staging of the A (d) tiles.
// ---------------------------------------------------------------------------
__global__ __launch_bounds__(128) void einsum_wmma_kernel(
    const float* __restrict__ d, const float* __restrict__ yT,
    float* __restrict__ out)
{
    __shared__ float sd[2][16 * LDS_STRIDE];

    const int tid  = threadIdx.x;
    const int lane = tid & 31;
    const int wave = tid >> 5;
    const int n0   = (blockIdx.x * 4 + wave) * 16;   // this wave's N tile

    // ---- load B fragments once: yT rows n0..n0+15, all K=256 ----
    // B(k,n): lane = n + 16*(k/2), reg = k%2  ->  per k-step kk (k = 4kk..4kk+3),
    // lane needs yT[(n0 + lane%16)*K + 4kk + 2*(lane/16) + {0,1}]  (contiguous pair)
    v2f bfrag[64];
    {
        const int   bn = n0 + (lane & 15);
        const int   bk = 2 * (lane >> 4);
        const float* bp = yT + (size_t)bn * K_ + bk;
        #pragma unroll
        for (int kk = 0; kk < 64; ++kk)
            bfrag[kk] = *(const v2f*)(bp + 4 * kk);
    }

    // A fragment addressing in LDS: A(m,k): lane = m + 16*(k/2), reg = k%2
    const int am = lane & 15;
    const int ak = 2 * (lane >> 4);

    const int mb0 = blockIdx.y * MB_PER_BLOCK;

    // ---- prologue: stage tile 0 into buffer 0 ----
    async_tile_load(d + (size_t)mb0 * 16 * K_, &sd[0][0], tid);
    wait_async_all();
    __syncthreads();

    for (int i = 0; i < MB_PER_BLOCK; ++i) {
        const int cur = i & 1;

        // ---- prefetch next tile into the other buffer (async, overlapped) ----
        const int inext = (i + 1 < MB_PER_BLOCK) ? (i + 1) : i;  // last iter: benign reload
        async_tile_load(d + (size_t)(mb0 + inext) * 16 * K_, &sd[1 - cur][0], tid);

        const float* sb = &sd[cur][0];

        // ---- K loop: 64 x v_wmma_f32_16x16x4_f32, fp32-exact accumulate ----
        // A fragments software-pipelined in 4 chunks of 16 (two buffers) to batch ds waits.
        v8f acc = {};
        v2f a[2][16];
        #pragma unroll
        for (int t = 0; t < 16; ++t)
            a[0][t] = *(const v2f*)(&sb[am * LDS_STRIDE + 4 * t + ak]);

        #pragma unroll
        for (int c = 0; c < 4; ++c) {
            if (c < 3) {
                #pragma unroll
                for (int t = 0; t < 16; ++t)
                    a[(c + 1) & 1][t] =
                        *(const v2f*)(&sb[am * LDS_STRIDE + 4 * (16 * (c + 1) + t) + ak]);
            }
            #pragma unroll
            for (int t = 0; t < 16; ++t)
                acc = __builtin_amdgcn_wmma_f32_16x16x4_f32(
                    /*neg_a=*/false, a[c & 1][t], /*neg_b=*/false, bfrag[16 * c + t],
                    /*c_mod=*/(short)0, acc, /*reuse_a=*/false, /*reuse_b=*/false);
        }

        // ---- write C: C(m,n): lane = n + 16*(m/8), reg = m%8 ----
        const int m0 = (mb0 + i) * 16 + 8 * (lane >> 4);
        float* op = out + (size_t)m0 * F_ + n0 + (lane & 15);
        #pragma unroll
        for (int v = 0; v < 8; ++v)
            op[(size_t)v * F_] = acc[v];

        // ---- drain our async writes, then sync the workgroup before swap ----
        wait_async_all();
        __syncthreads();
    }
}

// ---------------------------------------------------------------------------
extern "C" void kernel_launch(void* const* d_in, const int* in_sizes, int n_in,
                              void* d_out, int out_size, void* d_ws, size_t ws_size,
                              hipStream_t stream) {
    const float* x  = (const float*)d_in[0];   // (H, WG, F)
    const float* d  = (const float*)d_in[1];   // (H, WG, K)
    const float* W  = (const float*)d_in[2];   // (F, F)
    const int*   sx = (const int*)d_in[3];     // (K,)
    const int*   sy = (const int*)d_in[4];     // (K,)
    float* out = (float*)d_out;                // (H, WG, F)
    float* yT  = (float*)d_ws;                 // (F, K) transposed y, 256 KB

    station_gemm_kernel<<<dim3(K_), dim3(256), 0, stream>>>(x, W, sx, sy, yT);
    einsum_wmma_kernel<<<dim3(4, GRID_Y), dim3(128), 0, stream>>>(d, yT, out);
}